// RaZeRQuantizer_10170482557275
// MI455X (gfx1250) — compile-verified
//
#include <hip/hip_runtime.h>

// RaZeR FP4 group quantizer for gfx1250 (wave32).
// One wave = FOUR 128-float groups (one per 8-lane octet, 16 contiguous
// elements per lane). Per-group scalar setup amortized 4x; 3-stage
// ds_swizzle octet reductions. Candidate levels w[grid][16] are staged in
// LDS (bank-skewed, 16B-aligned) instead of spilling to scratch; the argmin
// grid is re-read with best-indexed ds_load_b128.

#define GS 128
#define WAVES_PER_BLOCK 4     // 128 threads/block (static LDS < 64KB)
#define GPW 4                 // groups per wave (one per 8-lane octet)
#define EPL 16                // elements per lane (GS*GPW/32)
#define LSTRIDE 68            // dwords per lane in LDS: 272B, 16B-aligned,
                              // bank skew 4*tid -> worst 2-way conflict

typedef float v4f __attribute__((ext_vector_type(4)));

// ds_swizzle_b32 group-of-32: offset15=0, [14:10]=xor, [9:5]=or, [4:0]=and.
template <int XOR>
__device__ __forceinline__ float swz(float v) {
  return __builtin_bit_cast(
      float, __builtin_amdgcn_ds_swizzle(__builtin_bit_cast(int, v),
                                         (XOR << 10) | 0x1F));
}

__device__ __forceinline__ void octred_minmax(float& mn, float& mx) {
  { float a = swz<4>(mn), b = swz<4>(mx); mn = fminf(mn, a); mx = fmaxf(mx, b); }
  { float a = swz<2>(mn), b = swz<2>(mx); mn = fminf(mn, a); mx = fmaxf(mx, b); }
  { float a = swz<1>(mn), b = swz<1>(mx); mn = fminf(mn, a); mx = fmaxf(mx, b); }
}

__device__ __forceinline__ void octred_sum4(float& e0, float& e1, float& e2,
                                            float& e3) {
  { float t0=swz<4>(e0), t1=swz<4>(e1), t2=swz<4>(e2), t3=swz<4>(e3);
    e0+=t0; e1+=t1; e2+=t2; e3+=t3; }
  { float t0=swz<2>(e0), t1=swz<2>(e1), t2=swz<2>(e2), t3=swz<2>(e3);
    e0+=t0; e1+=t1; e2+=t2; e3+=t3; }
  { float t0=swz<1>(e0), t1=swz<1>(e1), t2=swz<1>(e2), t3=swz<1>(e3);
    e0+=t0; e1+=t1; e2+=t2; e3+=t3; }
}

// Inner region of every grid is integers -4..4: clamp to [-4.5,4.5]
// (v_med3_f32) then round-to-nearest-even (v_rndne_f32).
__device__ __forceinline__ float rnd_inner(float xs) {
  return __builtin_rintf(__builtin_amdgcn_fmed3f(xs, -4.5f, 4.5f));
}

__global__ __launch_bounds__(WAVES_PER_BLOCK * 32)
void razer_quant_kernel(const float* __restrict__ x, float* __restrict__ out,
                        int ngroups) {
  __shared__ float wbuf[WAVES_PER_BLOCK * 32 * LSTRIDE];   // 34816 B

  const int wv = blockIdx.x * WAVES_PER_BLOCK + (threadIdx.x >> 5);
  const int tile = wv * GPW;              // first group of this wave's tile
  if (tile >= ngroups) return;            // wave-uniform guard
  const int lane = threadIdx.x & 31;
  // Lane owns contiguous elements [16*lane, 16*lane+16) of the 512-elt tile.
  const size_t base = (size_t)tile * GS + (size_t)lane * EPL;
  const int wb = (int)threadIdx.x * LSTRIDE;   // lane's LDS base (dwords)

  v4f c[4];
  c[0] = __builtin_nontemporal_load(reinterpret_cast<const v4f*>(x + base));
  c[1] = __builtin_nontemporal_load(reinterpret_cast<const v4f*>(x + base + 4));
  c[2] = __builtin_nontemporal_load(reinterpret_cast<const v4f*>(x + base + 8));
  c[3] = __builtin_nontemporal_load(reinterpret_cast<const v4f*>(x + base + 12));
  __builtin_prefetch(x + base + (size_t)(2 * 1024 * 1024 / 4), 0, 0);

  float mn = c[0][0], mx = c[0][0];
#pragma unroll
  for (int i = 0; i < 4; ++i)
#pragma unroll
    for (int j = 0; j < 4; ++j) {
      mn = fminf(mn, c[i][j]);
      mx = fmaxf(mx, c[i][j]);
    }
  octred_minmax(mn, mx);                  // octet = this lane's group
  const float rng = mx - mn;

  // Shared scales: grids {0,1} use range 24, grids {2,3} use range 28.
  const float sA = rng / 24.0f;           // exact IEEE div
  const float sB = rng / 28.0f;
  const float invA = 1.0f / sA;           // correctly rounded
  const float invB = 1.0f / sB;

  const float ZR2 = 12.0f / 28.0f, ZR2C = 1.0f - (12.0f / 28.0f);
  const float ZR3 = 16.0f / 28.0f, ZR3C = 1.0f - (16.0f / 28.0f);
  const float zpA = mx * 0.5f + mn * 0.5f;     // zr = 0.5 for grids 0,1
  const float zp2 = mx * ZR2 + mn * ZR2C;
  const float zp3 = mx * ZR3 + mn * ZR3C;
  const float rzpA = __builtin_rintf(zpA * invA);  // banker's rounding
  const float rzp2 = __builtin_rintf(zp2 * invB);
  const float rzp3 = __builtin_rintf(zp3 * invB);

  float err0 = 0.f, err1 = 0.f, err2 = 0.f, err3 = 0.f;
#pragma unroll
  for (int ch = 0; ch < 4; ++ch) {
    v4f wv0, wv1, wv2, wv3;
#pragma unroll
    for (int j = 0; j < 4; ++j) {
      const float xv = c[ch][j];

      // ---- grids 0,1: shared xs / inner round / {±5,±7} prefix ----
      const float xsA = __builtin_fmaf(xv, invA, -rzpA);
      float r01 = rnd_inner(xsA);
      r01 = (xsA > 5.f) ? 6.f : r01;
      r01 = (xsA > 7.f) ? 8.f : r01;
      r01 = (xsA <= -5.f) ? -6.f : r01;
      r01 = (xsA <= -7.f) ? -8.f : r01;
      float w0 = r01;                     // SM_POS tails {10,12} / {-12}
      w0 = (xsA > 9.f) ? 10.f : w0;
      w0 = (xsA > 11.f) ? 12.f : w0;
      w0 = (xsA <= -10.f) ? -12.f : w0;
      float w1 = r01;                     // SM_NEG tails {12} / {-10,-12}
      w1 = (xsA > 10.f) ? 12.f : w1;
      w1 = (xsA <= -9.f) ? -10.f : w1;
      w1 = (xsA <= -11.f) ? -12.f : w1;

      // ---- grid 2 (SR_POS) ----
      const float xs2 = __builtin_fmaf(xv, invB, -rzp2);
      float w2 = rnd_inner(xs2);
      w2 = (xs2 > 5.f) ? 6.f : w2;
      w2 = (xs2 > 7.f) ? 8.f : w2;
      w2 = (xs2 > 10.f) ? 12.f : w2;
      w2 = (xs2 > 14.f) ? 16.f : w2;
      w2 = (xs2 <= -5.f) ? -6.f : w2;
      w2 = (xs2 <= -7.f) ? -8.f : w2;
      w2 = (xs2 <= -10.f) ? -12.f : w2;

      // ---- grid 3 (SR_NEG) ----
      const float xs3 = __builtin_fmaf(xv, invB, -rzp3);
      float w3 = rnd_inner(xs3);
      w3 = (xs3 > 5.f) ? 6.f : w3;
      w3 = (xs3 > 7.f) ? 8.f : w3;
      w3 = (xs3 > 10.f) ? 12.f : w3;
      w3 = (xs3 <= -5.f) ? -6.f : w3;
      w3 = (xs3 <= -7.f) ? -8.f : w3;
      w3 = (xs3 <= -10.f) ? -12.f : w3;
      w3 = (xs3 <= -14.f) ? -16.f : w3;

      wv0[j] = w0; wv1[j] = w1; wv2[j] = w2; wv3[j] = w3;

      float d;
      d = w0 - xsA; err0 = __builtin_fmaf(d, d, err0);
      d = w1 - xsA; err1 = __builtin_fmaf(d, d, err1);
      d = w2 - xs2; err2 = __builtin_fmaf(d, d, err2);
      d = w3 - xs3; err3 = __builtin_fmaf(d, d, err3);
    }
    // Stage this chunk's 4 grids to LDS (ds_store_b128 x4, bank-skewed).
    *reinterpret_cast<v4f*>(&wbuf[wb + 0 * 16 + ch * 4]) = wv0;
    *reinterpret_cast<v4f*>(&wbuf[wb + 1 * 16 + ch * 4]) = wv1;
    *reinterpret_cast<v4f*>(&wbuf[wb + 2 * 16 + ch * 4]) = wv2;
    *reinterpret_cast<v4f*>(&wbuf[wb + 3 * 16 + ch * 4]) = wv3;
  }

  octred_sum4(err0, err1, err2, err3);
  const float fA = sA * sA, fB = sB * sB;
  err0 *= fA; err1 *= fA; err2 *= fB; err3 *= fB;

  // First-occurrence argmin over 4 scores (matches jnp.argmin).
  int best = 0;
  float bs = err0;
  if (err1 < bs) { bs = err1; best = 1; }
  if (err2 < bs) { bs = err2; best = 2; }
  if (err3 < bs) { bs = err3; best = 3; }

  const float scale_sel = (best >= 2) ? sB : sA;
  float rzp_sel = (best == 2) ? rzp2 : rzpA;
  rzp_sel = (best == 3) ? rzp3 : rzp_sel;

  // Fetch the winning grid's levels (per-lane dynamic index -> ds_load_b128),
  // dequant with the exact reference formula, stream out.
  const int rb = wb + best * 16;
#pragma unroll
  for (int ch = 0; ch < 4; ++ch) {
    const v4f ws = *reinterpret_cast<const v4f*>(&wbuf[rb + ch * 4]);
    v4f o;
#pragma unroll
    for (int j = 0; j < 4; ++j) o[j] = (ws[j] + rzp_sel) * scale_sel;
    __builtin_nontemporal_store(o,
                                reinterpret_cast<v4f*>(out + base + 4 * ch));
  }
}

extern "C" void kernel_launch(void* const* d_in, const int* in_sizes, int n_in,
                              void* d_out, int out_size, void* d_ws,
                              size_t ws_size, hipStream_t stream) {
  (void)n_in; (void)d_ws; (void)ws_size; (void)out_size;
  const float* x = (const float*)d_in[0];
  float* out = (float*)d_out;
  const int n = in_sizes[0];                       // 4096*4096
  const int ngroups = n / GS;                      // 131072 groups
  const int gpb = WAVES_PER_BLOCK * GPW;           // 16 groups per block
  const int blocks = (ngroups + gpb - 1) / gpb;    // 8192 blocks
  razer_quant_kernel<<<blocks, WAVES_PER_BLOCK * 32, 0, stream>>>(x, out,
                                                                  ngroups);
}